// NODE_18004502905504
// MI455X (gfx1250) — compile-verified
//
#include <hip/hip_runtime.h>

typedef float v2f __attribute__((ext_vector_type(2)));
typedef float v8f __attribute__((ext_vector_type(8)));

#define TREES   8
#define TDEPTH  6
#define NSEL    48     // TREES * TDEPTH
#define INDIM   64
#define LEAVES  64
#define WAVES   8
#define ITERS   4
#define ROWS_PER_BLOCK (16 * ITERS * WAVES)   // 512

__device__ __forceinline__ unsigned ballot32(bool p) {
    return __builtin_amdgcn_ballot_w32(p);
}

// Select m[s] for s in [0,8) using only constant indexing (cndmask tree, no scratch).
__device__ __forceinline__ unsigned pick8(const unsigned m[8], int s) {
    unsigned v0 = (s & 1) ? m[1] : m[0];
    unsigned v1 = (s & 1) ? m[3] : m[2];
    unsigned v2 = (s & 1) ? m[5] : m[4];
    unsigned v3 = (s & 1) ? m[7] : m[6];
    unsigned w0 = (s & 2) ? v1 : v0;
    unsigned w1 = (s & 2) ? v3 : v2;
    return (s & 4) ? w1 : w0;
}

__global__ __launch_bounds__(256) void node_forest_kernel(
    const float* __restrict__ x,          // [B,64]
    const float* __restrict__ W_sel,      // [8,6,64] == [48,64]
    const float* __restrict__ b_sel,      // [48]
    const float* __restrict__ leaf_values,// [8,64,2]
    const float* __restrict__ fc_w,       // [2,16]
    const float* __restrict__ fc_b,       // [2]
    float* __restrict__ out,              // [B,2]
    int Bsz)
{
    // B-matrix fragments, pre-paired: sWB[p*48+n] = (W_sel[n][2p], W_sel[n][2p+1])
    __shared__ v2f   sWB[32 * NSEL];          // 12 KB
    __shared__ v2f   sLeaf[TREES * LEAVES];   // 4 KB
    __shared__ float sBias[NSEL];
    __shared__ float sFcw[32];
    __shared__ float sFcb[2];

    const int tid = threadIdx.x;

    const v2f* Wv = (const v2f*)W_sel;        // Wv[n*32+p] = (W[n][2p], W[n][2p+1])
    for (int i = tid; i < 32 * NSEL; i += 256) {
        int p = i / NSEL, n = i - p * NSEL;
        sWB[i] = Wv[n * 32 + p];
    }
    const v2f* Lv = (const v2f*)leaf_values;
    for (int i = tid; i < TREES * LEAVES; i += 256) sLeaf[i] = Lv[i];
    if (tid < NSEL) sBias[tid] = b_sel[tid];
    if (tid < 32)   sFcw[tid]  = fc_w[tid];
    if (tid < 2)    sFcb[tid]  = fc_b[tid];
    __syncthreads();

    const int lane = tid & 31;
    const int wave = tid >> 5;
    const int mrow = lane & 15;   // M index (A) / N index (B,C)
    const int hi   = lane >> 4;   // 0: K+{0,1}, 1: K+{2,3}

    const float bb0 = sBias[mrow];
    const float bb1 = sBias[mrow + 16];
    const float bb2 = sBias[mrow + 32];

    const int  rho   = lane >> 1;       // my output row within the tile
    const int  jcomp = lane & 1;        // my output component
    const int  rsel  = rho & 7;         // accumulator component for my row
    const bool hiRow = (rho & 8) != 0;  // rows 8..15 live in ballot high half

    for (int it = 0; it < ITERS; ++it) {
        const long rowBase = (long)blockIdx.x * ROWS_PER_BLOCK + wave * (16 * ITERS) + it * 16;
        if (rowBase + 16 > Bsz) break;  // wave-uniform; EXEC stays all-ones

        // ---- load A fragments: lane L holds x[row=L%16][4k + 2*(L>=16) + {0,1}] ----
        const v2f* xv = (const v2f*)x + (rowBase + mrow) * 32 + hi;
        v2f afrag[16];
        #pragma unroll
        for (int k = 0; k < 16; ++k) afrag[k] = xv[2 * k];

        // ---- 16x48 logits via 48x V_WMMA_F32_16X16X4_F32 ----
        v8f acc0 = {0,0,0,0,0,0,0,0};
        v8f acc1 = {0,0,0,0,0,0,0,0};
        v8f acc2 = {0,0,0,0,0,0,0,0};
        #pragma unroll
        for (int k = 0; k < 16; ++k) {
            const v2f* wb = &sWB[(2 * k + hi) * NSEL + mrow];
            v2f b0 = wb[0];    // N-tile 0: n = mrow
            v2f b1 = wb[16];   // N-tile 1: n = mrow + 16
            v2f b2 = wb[32];   // N-tile 2: n = mrow + 32
            acc0 = __builtin_amdgcn_wmma_f32_16x16x4_f32(false, afrag[k], false, b0,
                                                         (short)0, acc0, false, false);
            acc1 = __builtin_amdgcn_wmma_f32_16x16x4_f32(false, afrag[k], false, b1,
                                                         (short)0, acc1, false, false);
            acc2 = __builtin_amdgcn_wmma_f32_16x16x4_f32(false, afrag[k], false, b2,
                                                         (short)0, acc2, false, false);
        }

        // ---- threshold + ballot: bit(row, n) with sigmoid(z)>0.5 <=> z>0 ----
        unsigned m0[8], m1[8], m2[8];
        #pragma unroll
        for (int r = 0; r < 8; ++r) {
            m0[r] = ballot32(acc0[r] + bb0 > 0.0f);
            m1[r] = ballot32(acc1[r] + bb1 > 0.0f);
            m2[r] = ballot32(acc2[r] + bb2 > 0.0f);
        }

        // ---- per-lane: row rho, component jcomp ----
        unsigned mm0 = pick8(m0, rsel);
        unsigned mm1 = pick8(m1, rsel);
        unsigned mm2 = pick8(m2, rsel);
        unsigned bits[3];
        bits[0] = hiRow ? (mm0 >> 16) : (mm0 & 0xffffu);
        bits[1] = hiRow ? (mm1 >> 16) : (mm1 & 0xffffu);
        bits[2] = hiRow ? (mm2 >> 16) : (mm2 & 0xffffu);

        float o = sFcb[jcomp];
        #pragma unroll
        for (int t = 0; t < TREES; ++t) {
            int idx = 0;
            #pragma unroll
            for (int d = 0; d < TDEPTH; ++d) {
                int n = TDEPTH * t + d;                 // compile-time constant
                idx = (idx << 1) | (int)((bits[n >> 4] >> (n & 15)) & 1u);
            }
            v2f lv = sLeaf[t * LEAVES + idx];
            o += sFcw[jcomp * 16 + 2 * t]     * lv.x
               + sFcw[jcomp * 16 + 2 * t + 1] * lv.y;
        }

        // row = rowBase + rho, comp jcomp  ->  flat offset rho*2 + jcomp == lane
        out[rowBase * 2 + lane] = o;
    }
}

extern "C" void kernel_launch(void* const* d_in, const int* in_sizes, int n_in,
                              void* d_out, int out_size, void* d_ws, size_t ws_size,
                              hipStream_t stream) {
    const float* x    = (const float*)d_in[0];
    const float* Wsel = (const float*)d_in[1];
    const float* bsel = (const float*)d_in[2];
    const float* leaf = (const float*)d_in[3];
    const float* fcw  = (const float*)d_in[4];
    const float* fcb  = (const float*)d_in[5];
    float* out = (float*)d_out;

    int Bsz = in_sizes[0] / INDIM;
    int blocks = (Bsz + ROWS_PER_BLOCK - 1) / ROWS_PER_BLOCK;
    node_forest_kernel<<<blocks, 256, 0, stream>>>(x, Wsel, bsel, leaf, fcw, fcb, out, Bsz);
}